// GNN_66322884984930
// MI455X (gfx1250) — compile-verified
//
#include <hip/hip_runtime.h>
#include <math.h>

// ---- problem constants ----
constexpr int kN   = 8000;           // nodes
constexpr int kSeq = 512;
constexpr int kD   = 513;            // D_EMB
constexpr int kDP  = 528;            // D padded to 16*33 (zeros -> exact GEMM)
constexpr int kNE  = 256000;         // edges
constexpr int kNP  = kN * kDP;

typedef __attribute__((ext_vector_type(2))) float v2f;
typedef __attribute__((ext_vector_type(8))) float v8f;

// emb[i, d] = src[i, d] (d<512) ; one-hot pert col at d==512 ; 0 pad beyond
__global__ void k_build_emb(const float* __restrict__ src,
                            const int* __restrict__ pert,
                            float* __restrict__ emb) {
    int idx = blockIdx.x * blockDim.x + threadIdx.x;
    if (idx >= kNP) return;
    int i = idx / kDP, d = idx % kDP;
    float v = 0.f;
    if (d < kSeq)       v = src[i * kSeq + d];
    else if (d == kSeq) v = (i == *pert) ? 1.f : 0.f;
    emb[idx] = v;
}

__global__ void k_init_deg(float* __restrict__ deg) {
    int i = blockIdx.x * blockDim.x + threadIdx.x;
    if (i < kN) deg[i] = 1.0f;              // self-loop weight
}

__global__ void k_edge_deg(const int* __restrict__ ei,
                           const float* __restrict__ ew,
                           float* __restrict__ deg) {
    int e = blockIdx.x * blockDim.x + threadIdx.x;
    if (e >= kNE) return;
    atomicAdd(&deg[ei[kNE + e]], ew[e]);    // segment_sum over col
}

__global__ void k_dis(const float* __restrict__ deg, float* __restrict__ dis) {
    int i = blockIdx.x * blockDim.x + threadIdx.x;
    if (i < kN) dis[i] = rsqrtf(deg[i]);    // deg >= 1 always
}

// agg = (dis[i]^2) * emb_in[i]  (self-loop term; also zero-inits padded cols)
__global__ void k_selfloop(const float* __restrict__ emb_in,
                           const float* __restrict__ dis,
                           float* __restrict__ agg) {
    int idx = blockIdx.x * blockDim.x + threadIdx.x;
    if (idx >= kNP) return;
    int i = idx / kDP;
    float s = dis[i];
    agg[idx] = s * s * emb_in[idx];
}

// one wave per edge: agg[col] += dis[row]*w*dis[col] * emb_in[row]
__global__ void k_edge_scatter(const int* __restrict__ ei,
                               const float* __restrict__ ew,
                               const float* __restrict__ dis,
                               const float* __restrict__ emb_in,
                               float* __restrict__ agg) {
    int gid  = blockIdx.x * blockDim.x + threadIdx.x;
    int e    = gid >> 5;
    int lane = gid & 31;
    if (e >= kNE) return;
    int r = ei[e];
    int c = ei[kNE + e];
    float nm = dis[r] * ew[e] * dis[c];
    const float* src = emb_in + r * kDP;
    float* dst = agg + c * kDP;
    for (int d = lane; d < kD; d += 32)
        atomicAdd(&dst[d], nm * src[d]);
}

// pad both layer weights [513,513] -> [528,528] with zeros
__global__ void k_pad_w(const float* __restrict__ gW, float* __restrict__ Wp) {
    int idx = blockIdx.x * blockDim.x + threadIdx.x;
    if (idx >= 2 * kDP * kDP) return;
    int l = idx / (kDP * kDP);
    int r = idx % (kDP * kDP);
    int n = r / kDP, k = r % kDP;
    Wp[idx] = (n < kD && k < kD) ? gW[l * kD * kD + n * kD + k] : 0.f;
}

// out[M=8000, N=528] = A[M, K=528] @ W^T + bias  (W padded row-major [n][k])
// one wave per 64x16 output block: 4 M-tiles share one B fragment per K step
// (V_WMMA_F32_16X16X4_F32; 10 bytes of fragment loads per WMMA vs 16 naive)
__global__ void __launch_bounds__(256)
k_gemm_wmma(const float* __restrict__ A,
            const float* __restrict__ W,
            const float* __restrict__ bias,
            float* __restrict__ out, int relu) {
    constexpr int TMG = kN / 64;   // 125 M-groups of 4 tiles
    constexpr int TN  = kDP / 16;  // 33
    int wave = threadIdx.x >> 5;
    int lane = threadIdx.x & 31;
    int blk  = blockIdx.x * 8 + wave;
    if (blk >= TMG * TN) return;               // wave-uniform exit (EXEC all-1)
    int tmg = blk / TN, tn = blk % TN;
    int m0 = tmg * 64, n0 = tn * 16;
    int g   = lane >> 4;                        // lane group
    int lid = lane & 15;
    int kb  = g * 2;
    // A frag: lanes 0-15 hold M rows {K=kb,kb+1}; B frag: lanes hold N cols
    const float* pa0 = A + (m0      + lid) * kDP + kb;
    const float* pa1 = A + (m0 + 16 + lid) * kDP + kb;
    const float* pa2 = A + (m0 + 32 + lid) * kDP + kb;
    const float* pa3 = A + (m0 + 48 + lid) * kDP + kb;
    const float* pb  = W + (n0 + lid) * kDP + kb;   // B[k][n] = W[n][k]
    v8f c0 = {0.f,0.f,0.f,0.f,0.f,0.f,0.f,0.f};
    v8f c1 = c0, c2 = c0, c3 = c0;
#pragma unroll 4
    for (int k = 0; k < kDP; k += 4) {
        v2f b  = *reinterpret_cast<const v2f*>(pb  + k);
        v2f a0 = *reinterpret_cast<const v2f*>(pa0 + k);
        v2f a1 = *reinterpret_cast<const v2f*>(pa1 + k);
        v2f a2 = *reinterpret_cast<const v2f*>(pa2 + k);
        v2f a3 = *reinterpret_cast<const v2f*>(pa3 + k);
        c0 = __builtin_amdgcn_wmma_f32_16x16x4_f32(false, a0, false, b,
                                                   (short)0, c0, false, false);
        c1 = __builtin_amdgcn_wmma_f32_16x16x4_f32(false, a1, false, b,
                                                   (short)0, c1, false, false);
        c2 = __builtin_amdgcn_wmma_f32_16x16x4_f32(false, a2, false, b,
                                                   (short)0, c2, false, false);
        c3 = __builtin_amdgcn_wmma_f32_16x16x4_f32(false, a3, false, b,
                                                   (short)0, c3, false, false);
    }
    int n = n0 + lid;
    float bv = (n < kD) ? bias[n] : 0.f;
    // C/D layout: VGPR v -> M = v (lanes 0-15) / v+8 (lanes 16-31)
#pragma unroll
    for (int v = 0; v < 8; ++v) {
        int mrow = v + g * 8;
        float x0 = c0[v] + bv;
        float x1 = c1[v] + bv;
        float x2 = c2[v] + bv;
        float x3 = c3[v] + bv;
        if (relu) {
            x0 = fmaxf(x0, 0.f); x1 = fmaxf(x1, 0.f);
            x2 = fmaxf(x2, 0.f); x3 = fmaxf(x3, 0.f);
        }
        out[(m0      + mrow) * kDP + n] = x0;
        out[(m0 + 16 + mrow) * kDP + n] = x1;
        out[(m0 + 32 + mrow) * kDP + n] = x2;
        out[(m0 + 48 + mrow) * kDP + n] = x3;
    }
}

// u[d] = sum_j (scale * q_pert[j]) * Wk[j,d];  (bk-term cancels in softmax)
__global__ void k_attn_prep(const float* __restrict__ emb,
                            const float* __restrict__ inw,
                            const float* __restrict__ inb,
                            const int* __restrict__ pert,
                            float* __restrict__ u) {
    __shared__ float qs[kD];
    int tid = threadIdx.x;
    const float scale = rsqrtf((float)kD);
    const float* ep = emb + (*pert) * kDP;
    for (int j = tid; j < kD; j += blockDim.x) {
        float acc = inb[j];                         // bq
        const float* wr = inw + j * kD;             // Wq row j
        for (int d = 0; d < kD; ++d) acc += ep[d] * wr[d];
        qs[j] = acc * scale;
    }
    __syncthreads();
    for (int d = tid; d < kD; d += blockDim.x) {
        float acc = 0.f;
        for (int j = 0; j < kD; ++j)
            acc += qs[j] * inw[(kD + j) * kD + d];  // Wk rows start at kD
        u[d] = acc;
    }
}

// scores[i] = emb[i] . u   (one wave per row)
__global__ void k_scores(const float* __restrict__ emb,
                         const float* __restrict__ u,
                         float* __restrict__ sc) {
    int wave = threadIdx.x >> 5;
    int lane = threadIdx.x & 31;
    int row  = blockIdx.x * 8 + wave;
    if (row >= kN) return;
    const float* er = emb + row * kDP;
    float v = 0.f;
    for (int d = lane; d < kD; d += 32) v += er[d] * u[d];
    for (int off = 16; off > 0; off >>= 1) v += __shfl_down(v, off, 32);
    if (lane == 0) sc[row] = v;
}

// in-place softmax over kN scores (single block)
__global__ void k_softmax(float* __restrict__ sc) {
    __shared__ float red[1024];
    int tid = threadIdx.x;
    float m = -3.402823466e38f;
    for (int i = tid; i < kN; i += 1024) m = fmaxf(m, sc[i]);
    red[tid] = m; __syncthreads();
    for (int s = 512; s > 0; s >>= 1) {
        if (tid < s) red[tid] = fmaxf(red[tid], red[tid + s]);
        __syncthreads();
    }
    float mx = red[0]; __syncthreads();
    float sm = 0.f;
    for (int i = tid; i < kN; i += 1024) sm += expf(sc[i] - mx);
    red[tid] = sm; __syncthreads();
    for (int s = 512; s > 0; s >>= 1) {
        if (tid < s) red[tid] += red[tid + s];
        __syncthreads();
    }
    float inv = 1.f / red[0];
    for (int i = tid; i < kN; i += 1024) sc[i] = expf(sc[i] - mx) * inv;
}

// cell[d] = sum_i w[i] * emb[i,d]
__global__ void k_cell(const float* __restrict__ emb,
                       const float* __restrict__ w,
                       float* __restrict__ cell) {
    int d = blockIdx.x * blockDim.x + threadIdx.x;
    if (d >= kD) return;
    float acc = 0.f;
    for (int i = 0; i < kN; ++i) acc += w[i] * emb[i * kDP + d];
    cell[d] = acc;
}

// fused 4-layer decoder MLP, one block of 512
__global__ void k_decoder(const float* __restrict__ cell,
                          const float* __restrict__ W1, const float* __restrict__ b1,
                          const float* __restrict__ W2, const float* __restrict__ b2,
                          const float* __restrict__ W3, const float* __restrict__ b3,
                          const float* __restrict__ oW, const float* __restrict__ ob,
                          float* __restrict__ out) {
    __shared__ float cs[kD];
    __shared__ float h1[256];
    __shared__ float h2[256];
    __shared__ float h3[512];
    int tid = threadIdx.x;
    for (int d = tid; d < kD; d += 512) cs[d] = cell[d];
    __syncthreads();
    if (tid < 256) {
        float acc = b1[tid];
        const float* wr = W1 + tid * kD;
        for (int d = 0; d < kD; ++d) acc += wr[d] * cs[d];
        h1[tid] = fmaxf(acc, 0.f);
    }
    __syncthreads();
    if (tid < 256) {
        float acc = b2[tid];
        const float* wr = W2 + tid * 256;
        for (int k = 0; k < 256; ++k) acc += wr[k] * h1[k];
        h2[tid] = fmaxf(acc, 0.f);
    }
    __syncthreads();
    {
        float acc = b3[tid];
        const float* wr = W3 + tid * 256;
        for (int k = 0; k < 256; ++k) acc += wr[k] * h2[k];
        h3[tid] = fmaxf(acc, 0.f);
    }
    __syncthreads();
    {
        float acc = ob[tid];
        const float* wr = oW + tid * 512;
        for (int k = 0; k < 512; ++k) acc += wr[k] * h3[k];
        out[tid] = acc;
    }
}

extern "C" void kernel_launch(void* const* d_in, const int* in_sizes, int n_in,
                              void* d_out, int out_size, void* d_ws, size_t ws_size,
                              hipStream_t stream) {
    const float* src = (const float*)d_in[0];
    const int*   ei  = (const int*)d_in[1];
    const float* ew  = (const float*)d_in[2];
    const float* gW  = (const float*)d_in[3];
    const float* gb  = (const float*)d_in[4];
    const float* inw = (const float*)d_in[5];
    const float* inb = (const float*)d_in[6];
    // d_in[7], d_in[8] (out_proj) are dead in the reference
    const float* dW1 = (const float*)d_in[9];
    const float* db1 = (const float*)d_in[10];
    const float* dW2 = (const float*)d_in[11];
    const float* db2 = (const float*)d_in[12];
    const float* dW3 = (const float*)d_in[13];
    const float* db3 = (const float*)d_in[14];
    const float* oW  = (const float*)d_in[15];
    const float* ob  = (const float*)d_in[16];
    const int*   pert = (const int*)d_in[17];
    float* out = (float*)d_out;

    float* ws   = (float*)d_ws;
    float* bufA = ws;                 // emb0, later emb_final
    float* bufB = bufA + kNP;         // agg (both layers)
    float* bufC = bufB + kNP;         // emb1
    float* Wpad = bufC + kNP;         // 2 * 528 * 528
    float* deg  = Wpad + 2 * kDP * kDP;
    float* dis  = deg + kN;
    float* sc   = dis + kN;
    float* u    = sc + kN;
    float* cell = u + kDP;

    k_build_emb<<<(kNP + 255) / 256, 256, 0, stream>>>(src, pert, bufA);
    k_init_deg<<<(kN + 255) / 256, 256, 0, stream>>>(deg);
    k_edge_deg<<<(kNE + 255) / 256, 256, 0, stream>>>(ei, ew, deg);
    k_dis<<<(kN + 255) / 256, 256, 0, stream>>>(deg, dis);
    k_pad_w<<<(2 * kDP * kDP + 255) / 256, 256, 0, stream>>>(gW, Wpad);

    const int WAVES = (kN / 64) * (kDP / 16);   // 125 * 33 = 4125
    const int GB = (WAVES + 7) / 8;

    // layer 0: agg -> emb1 (ReLU)
    k_selfloop<<<(kNP + 255) / 256, 256, 0, stream>>>(bufA, dis, bufB);
    k_edge_scatter<<<kNE / 8, 256, 0, stream>>>(ei, ew, dis, bufA, bufB);
    k_gemm_wmma<<<GB, 256, 0, stream>>>(bufB, Wpad, gb, bufC, 1);
    // layer 1: agg -> emb_final (no ReLU)
    k_selfloop<<<(kNP + 255) / 256, 256, 0, stream>>>(bufC, dis, bufB);
    k_edge_scatter<<<kNE / 8, 256, 0, stream>>>(ei, ew, dis, bufC, bufB);
    k_gemm_wmma<<<GB, 256, 0, stream>>>(bufB, Wpad + kDP * kDP, gb + kD, bufA, 0);

    // folded single-row attention
    k_attn_prep<<<1, 256, 0, stream>>>(bufA, inw, inb, pert, u);
    k_scores<<<kN / 8, 256, 0, stream>>>(bufA, u, sc);
    k_softmax<<<1, 1024, 0, stream>>>(sc);
    k_cell<<<(kD + 255) / 256, 256, 0, stream>>>(bufA, sc, cell);
    k_decoder<<<1, 512, 0, stream>>>(cell, dW1, db1, dW2, db2, dW3, db3, oW, ob, out);

    (void)in_sizes; (void)n_in; (void)out_size; (void)ws_size;
}